// net_86440511800198
// MI455X (gfx1250) — compile-verified
//
#include <hip/hip_runtime.h>

typedef __attribute__((ext_vector_type(2))) float v2f;
typedef __attribute__((ext_vector_type(8))) float v8f;

#define Bn   16
#define Cn   3
#define Hn   256
#define Wn   256
#define Kk   9
#define On   64
#define CK   27   // C*K contraction length
#define CKP  28   // padded to multiple of 4 for 16x16x4 WMMA
#define SROW 272  // LDS pixel stride: 272*2 mod 64 = 32 -> halves hit disjoint banks

__device__ __forceinline__ float samp(const float* __restrict__ p, int y, int xx) {
    if (y < 0 || y >= Hn || xx < 0 || xx >= Wn) return 0.0f;
    return p[y * Wn + xx];
}

__global__ __launch_bounds__(256)
void dcn_fused_kernel(const float* __restrict__ x,
                      const float* __restrict__ w_offset,
                      const float* __restrict__ b_offset,
                      const float* __restrict__ w_mask,
                      const float* __restrict__ b_mask,
                      const float* __restrict__ w_conv,
                      float* __restrict__ out)
{
    __shared__ float sS[CKP * SROW];  // sampled*mask: 28 rows x 272 px (30464 B)
    __shared__ float sW[On * CKP];    // w_conv zero-padded 64 x 28     (7168 B)

    const int tid  = threadIdx.x;
    const int b    = blockIdx.x >> 8;
    const int h    = blockIdx.x & 255;
    const int wpix = tid;

    // stage w_conv into LDS, zero-padding ck=27
    for (int i = tid; i < On * CKP; i += 256) {
        const int o = i / CKP, ck = i % CKP;
        sW[i] = (ck < CK) ? w_conv[o * CK + ck] : 0.0f;
    }

    // ---------------- phase 1: per-pixel offsets/mask/sampling ----------------
    const float* xb = x + (size_t)b * Cn * Hn * Wn;

    float patch[Cn][3][3];
#pragma unroll
    for (int c = 0; c < Cn; ++c)
#pragma unroll
        for (int i = 0; i < 3; ++i) {
            const int hh = h + i - 1;
#pragma unroll
            for (int j = 0; j < 3; ++j) {
                const int ww = wpix + j - 1;
                const bool ok = (hh >= 0) & (hh < Hn) & (ww >= 0) & (ww < Wn);
                patch[c][i][j] = ok ? xb[((size_t)c * Hn + hh) * Wn + ww] : 0.0f;
            }
        }

    float offv[2 * Kk];
#pragma unroll
    for (int t = 0; t < 2 * Kk; ++t) {
        float acc = b_offset[t];
#pragma unroll
        for (int c = 0; c < Cn; ++c)
#pragma unroll
            for (int i = 0; i < 3; ++i)
#pragma unroll
                for (int j = 0; j < 3; ++j)
                    acc = fmaf(w_offset[((t * Cn + c) * 3 + i) * 3 + j],
                               patch[c][i][j], acc);
        offv[t] = acc;
    }

    float mk[Kk];
#pragma unroll
    for (int t = 0; t < Kk; ++t) {
        float acc = b_mask[t];
#pragma unroll
        for (int c = 0; c < Cn; ++c)
#pragma unroll
            for (int i = 0; i < 3; ++i)
#pragma unroll
                for (int j = 0; j < 3; ++j)
                    acc = fmaf(w_mask[((t * Cn + c) * 3 + i) * 3 + j],
                               patch[c][i][j], acc);
        mk[t] = 1.0f / (1.0f + __expf(-acc));
    }

#pragma unroll
    for (int k = 0; k < Kk; ++k) {
        const float dy = offv[2 * k + 0];
        const float dx = offv[2 * k + 1];
        const float py = (float)h    + (float)(k / 3) - 1.0f + dy;
        const float px = (float)wpix + (float)(k % 3) - 1.0f + dx;
        const float y0f = floorf(py), x0f = floorf(px);
        const float wy = py - y0f, wx = px - x0f;
        const int y0 = (int)y0f, x0 = (int)x0f;
        const float w00 = (1.0f - wy) * (1.0f - wx);
        const float w01 = (1.0f - wy) * wx;
        const float w10 = wy * (1.0f - wx);
        const float w11 = wy * wx;
#pragma unroll
        for (int c = 0; c < Cn; ++c) {
            const float* xc = xb + (size_t)c * Hn * Wn;
            float v = w00 * samp(xc, y0,     x0)
                    + w01 * samp(xc, y0,     x0 + 1)
                    + w10 * samp(xc, y0 + 1, x0)
                    + w11 * samp(xc, y0 + 1, x0 + 1);
            sS[(c * Kk + k) * SROW + wpix] = v * mk[k];
        }
    }
    sS[CK * SROW + wpix] = 0.0f;  // zero-pad ck=27 row

    __syncthreads();

    // ---------------- phase 2: WMMA GEMM  out[64 x 256] = W[64 x 28] * S[28 x 256] --
    const int lane  = tid & 31;
    const int ln15  = lane & 15;
    const int khalf = (lane < 16) ? 0 : 2;  // A/B: VGPRj holds K = khalf + j

    // preload A (w_conv) in 16x4 fp32 A-matrix layout: 4 o-tiles x 7 K-steps
    v2f areg[4][7];
#pragma unroll
    for (int ot = 0; ot < 4; ++ot)
#pragma unroll
        for (int s = 0; s < 7; ++s) {
            const float* wp = &sW[(ot * 16 + ln15) * CKP + 4 * s + khalf];
            v2f a; a[0] = wp[0]; a[1] = wp[1];
            areg[ot][s] = a;
        }

    const int wave = tid >> 5;
#pragma unroll
    for (int g = 0; g < 2; ++g) {
        const int pixbase = wave * 32 + g * 16;
        v8f acc0 = {}, acc1 = {}, acc2 = {}, acc3 = {};
#pragma unroll
        for (int s = 0; s < 7; ++s) {
            const float* bp = &sS[(4 * s + khalf) * SROW + pixbase + ln15];
            v2f bm; bm[0] = bp[0]; bm[1] = bp[SROW];
            acc0 = __builtin_amdgcn_wmma_f32_16x16x4_f32(false, areg[0][s], false, bm,
                                                         (short)0, acc0, false, false);
            acc1 = __builtin_amdgcn_wmma_f32_16x16x4_f32(false, areg[1][s], false, bm,
                                                         (short)0, acc1, false, false);
            acc2 = __builtin_amdgcn_wmma_f32_16x16x4_f32(false, areg[2][s], false, bm,
                                                         (short)0, acc2, false, false);
            acc3 = __builtin_amdgcn_wmma_f32_16x16x4_f32(false, areg[3][s], false, bm,
                                                         (short)0, acc3, false, false);
        }

        // D layout: VGPR r -> M = r (lanes 0-15) / M = 8+r (lanes 16-31), N = lane&15
        const int mbase = (lane < 16) ? 0 : 8;
        const int pix   = pixbase + ln15;
        float* ob = out + (((size_t)b * On) * Hn + h) * Wn + pix;
#pragma unroll
        for (int r = 0; r < 8; ++r) {
            ob[(size_t)(0 * 16 + mbase + r) * Hn * Wn] = acc0[r];
            ob[(size_t)(1 * 16 + mbase + r) * Hn * Wn] = acc1[r];
            ob[(size_t)(2 * 16 + mbase + r) * Hn * Wn] = acc2[r];
            ob[(size_t)(3 * 16 + mbase + r) * Hn * Wn] = acc3[r];
        }
    }
}

extern "C" void kernel_launch(void* const* d_in, const int* in_sizes, int n_in,
                              void* d_out, int out_size, void* d_ws, size_t ws_size,
                              hipStream_t stream) {
    const float* x        = (const float*)d_in[0];
    const float* w_offset = (const float*)d_in[1];
    const float* b_offset = (const float*)d_in[2];
    const float* w_mask   = (const float*)d_in[3];
    const float* b_mask   = (const float*)d_in[4];
    const float* w_conv   = (const float*)d_in[5];
    float* out = (float*)d_out;

    dim3 grid(Bn * Hn);   // 4096 blocks: one per (batch, image row)
    dim3 block(256);      // 256 pixels = 8 wave32
    dcn_fused_kernel<<<grid, block, 0, stream>>>(x, w_offset, b_offset,
                                                 w_mask, b_mask, w_conv, out);
}